// GATFeaturePropagation_74431783240401
// MI455X (gfx1250) — compile-verified
//
#include <hip/hip_runtime.h>
#include <hip/hip_bf16.h>

// GAT feature propagation for MI455X (gfx1250, wave32).
// Pipeline:
//   K1: xt = x @ W            (fp32 WMMA, V_WMMA_F32_16X16X4_F32)
//   K2: scoreL/scoreR[n,h]    (per-node attention dot products, wave32 shuffle reduce)
//   K3: per-node softmax(4 heads) + gather-weighted sum of xt[col[n]]
//
// N=50000, IN_CH=256, HEADS=4, OUT_CH=64 -> W is 256x256, xt is N x 256.

typedef __attribute__((ext_vector_type(2))) float v2f;
typedef __attribute__((ext_vector_type(8))) float v8f;

#define IN_CH   256
#define HCOLS   256   // HEADS * OUT_CH
#define OUT_CH  64
#define HEADS   4

// ---------------------------------------------------------------------------
// Kernel 1: xt = x @ W with V_WMMA_F32_16X16X4_F32.
// Block = 256 threads = 8 waves. Block b owns output rows [16b, 16b+16).
// Wave w owns output cols [32w, 32w+16) and [32w+16, 32w+32)  (8*32 = 256 cols).
// K-loop: 256 / 4 = 64 WMMA steps per accumulator.
//
// VGPR layouts (ISA 7.12.2, 32-bit, wave32):
//   A 16x4 : VGPR0 = A[m, k+2*hi], VGPR1 = A[m, k+2*hi+1], m=lane&15, hi=lane>>4
//   B 4x16 : VGPR0 = B[k+2*hi, n], VGPR1 = B[k+2*hi+1, n], n=lane&15
//   C 16x16: VGPR v = C[v + 8*hi, n]
// ---------------------------------------------------------------------------
__global__ __launch_bounds__(256) void gat_gemm_wmma(
    const float* __restrict__ x,   // [n_rows, 256]
    const float* __restrict__ w,   // [256, 256] row-major (k, col)
    float* __restrict__ xt,        // [n_rows, 256]
    int n_rows)
{
  const int lane = threadIdx.x & 31;
  const int wave = threadIdx.x >> 5;
  const int row0 = blockIdx.x * 16;
  const int col0 = wave * 32;
  const int m    = lane & 15;
  const int hi   = lane >> 4;          // 0: lanes 0-15, 1: lanes 16-31

  // Clamp A-row reads for a possible ragged last tile (N=50000 is exact: 3125 tiles).
  int arow_idx = row0 + m;
  if (arow_idx >= n_rows) arow_idx = n_rows - 1;

  const float* arow  = x + (size_t)arow_idx * IN_CH + 2 * hi;       // A[m, k+2hi ...]
  const float* bbase = w + (size_t)(2 * hi) * HCOLS + col0 + m;     // B[k+2hi, col0+m]

  v8f c0 = {};
  v8f c1 = {};

  for (int k = 0; k < IN_CH; k += 4) {
    // A fragment: two consecutive K values for this half-wave.
    v2f a;
    a.x = arow[k];
    a.y = arow[k + 1];

    // B fragments for the two 16-wide column tiles.
    const float* bp = bbase + (size_t)k * HCOLS;
    v2f b0, b1;
    b0.x = bp[0];
    b0.y = bp[HCOLS];
    b1.x = bp[16];
    b1.y = bp[16 + HCOLS];

    // Prefetch next A cacheline (lowers to global_prefetch_b8 on gfx1250).
    __builtin_prefetch(arow + k + 4, 0, 0);

    // D = A*B + C   (8 args: neg_a, A, neg_b, B, c_mod, C, reuse_a, reuse_b)
    c0 = __builtin_amdgcn_wmma_f32_16x16x4_f32(false, a, false, b0,
                                               (short)0, c0, false, false);
    c1 = __builtin_amdgcn_wmma_f32_16x16x4_f32(false, a, false, b1,
                                               (short)0, c1, false, false);
  }

  // Store: VGPR v holds row (v + 8*hi), col (col0 + m) / (col0 + 16 + m).
  const int orow0 = row0 + 8 * hi;
  float* obase = xt + (size_t)orow0 * HCOLS + col0 + m;
#pragma unroll
  for (int v = 0; v < 8; ++v) {
    if (orow0 + v < n_rows) {
      obase[(size_t)v * HCOLS]      = c0[v];
      obase[(size_t)v * HCOLS + 16] = c1[v];
    }
  }
}

// ---------------------------------------------------------------------------
// Kernel 2: per-node attention scores.
//   scoreL[n,h] = sum_d xt[n, h*64+d] * attn_l[h*64+d]   (attn flat-aligned with row)
// One wave per node; lane covers 8 contiguous elements; head h == lanes 8h..8h+7,
// so an xor-shuffle reduce over {1,2,4} yields the per-head dot product.
// ---------------------------------------------------------------------------
__global__ __launch_bounds__(256) void gat_scores(
    const float* __restrict__ xt,
    const float* __restrict__ attn_l,   // [256] flat
    const float* __restrict__ attn_r,   // [256] flat
    float* __restrict__ sL,             // [n, 4]
    float* __restrict__ sR,             // [n, 4]
    int n)
{
  const int gwave = (int)((blockIdx.x * blockDim.x + threadIdx.x) >> 5);
  const int lane  = threadIdx.x & 31;
  if (gwave >= n) return;

  const float* xr = xt + (size_t)gwave * HCOLS + lane * 8;
  const float* al = attn_l + lane * 8;
  const float* ar = attn_r + lane * 8;

  float pl = 0.f, pr = 0.f;
#pragma unroll
  for (int i = 0; i < 8; ++i) {
    const float v = xr[i];
    pl += v * al[i];
    pr += v * ar[i];
  }
#pragma unroll
  for (int off = 1; off < 8; off <<= 1) {
    pl += __shfl_xor(pl, off, 32);
    pr += __shfl_xor(pr, off, 32);
  }
  if ((lane & 7) == 0) {
    const int h = lane >> 3;
    sL[(size_t)gwave * HEADS + h] = pl;
    sR[(size_t)gwave * HEADS + h] = pr;
  }
}

// ---------------------------------------------------------------------------
// Kernel 3: only edges e < N matter (reference uses alpha[:N]).
//   s[h]   = leaky_relu(scoreL[row[n],h] + scoreR[col[n],h], 0.01)
//   alpha  = softmax_h(s)
//   tmp[d] = sum_h alpha[h] * xt[col[n], h*64+d]
//   out[n, h*64+d] = tmp[d]  for all h   (broadcast across heads)
// 64 threads per node (one per d); grid covers N*64 threads.
// ---------------------------------------------------------------------------
__global__ __launch_bounds__(256) void gat_gather(
    const float* __restrict__ xt,
    const long long* __restrict__ edge,  // [2, E] int64
    const float* __restrict__ sL,
    const float* __restrict__ sR,
    float* __restrict__ out,             // [n, 256]
    int n, long long E)
{
  const long long idx = (long long)blockIdx.x * blockDim.x + threadIdx.x;
  const int node = (int)(idx >> 6);
  const int d    = (int)(idx & 63);
  if (node >= n) return;

  const int r = (int)edge[node];         // row[node]
  const int c = (int)edge[E + node];     // col[node]

  float s[HEADS];
  float mx = -3.402823466e+38f;
#pragma unroll
  for (int h = 0; h < HEADS; ++h) {
    float v = sL[(size_t)r * HEADS + h] + sR[(size_t)c * HEADS + h];
    v = (v > 0.f) ? v : 0.01f * v;       // leaky_relu(0.01)
    s[h] = v;
    mx = fmaxf(mx, v);
  }
  float sum = 0.f;
#pragma unroll
  for (int h = 0; h < HEADS; ++h) {
    s[h] = __expf(s[h] - mx);
    sum += s[h];
  }
  const float inv = 1.0f / sum;

  const float* xc = xt + (size_t)c * HCOLS + d;
  float tmp = 0.f;
#pragma unroll
  for (int h = 0; h < HEADS; ++h)
    tmp += s[h] * inv * xc[h * OUT_CH];

  float* o = out + (size_t)node * HCOLS + d;
#pragma unroll
  for (int h = 0; h < HEADS; ++h)
    o[h * OUT_CH] = tmp;
}

// ---------------------------------------------------------------------------
// Launch. Inputs (setup_inputs order): x, edge_index, weight, attn_l, attn_r.
// Workspace layout: xt [N*256 f32] | scoreL [N*4 f32] | scoreR [N*4 f32].
// ---------------------------------------------------------------------------
extern "C" void kernel_launch(void* const* d_in, const int* in_sizes, int n_in,
                              void* d_out, int out_size, void* d_ws, size_t ws_size,
                              hipStream_t stream) {
  const float*     x    = (const float*)d_in[0];
  const long long* edge = (const long long*)d_in[1];   // int64
  const float*     w    = (const float*)d_in[2];
  const float*     al   = (const float*)d_in[3];
  const float*     ar   = (const float*)d_in[4];
  float*           out  = (float*)d_out;

  const int       N = in_sizes[0] / IN_CH;             // 50000
  const long long E = (long long)in_sizes[1] / 2;      // 800000

  float* xt = (float*)d_ws;
  float* sL = xt + (size_t)N * HCOLS;
  float* sR = sL + (size_t)N * HEADS;

  const int rowTiles = (N + 15) / 16;                  // 3125
  gat_gemm_wmma<<<rowTiles, 256, 0, stream>>>(x, w, xt, N);

  const int scoreBlocks = (N + 7) / 8;                 // 8 waves per block
  gat_scores<<<scoreBlocks, 256, 0, stream>>>(xt, al, ar, sL, sR, N);

  const long long gthreads = (long long)N * OUT_CH;
  const int gatherBlocks = (int)((gthreads + 255) / 256);
  gat_gather<<<gatherBlocks, 256, 0, stream>>>(xt, edge, sL, sR, out, N, E);
}